// TransFunc_10831907521128
// MI455X (gfx1250) — compile-verified
//
#include <hip/hip_runtime.h>
#include <stdint.h>

typedef __attribute__((ext_vector_type(2))) float v2f;
typedef __attribute__((ext_vector_type(4))) float f4v;
typedef __attribute__((ext_vector_type(8))) float v8f;

#define N6   524288
#define M5   131072
#define M4   32768
#define CIN  256
#define COUT 512
#define NP5  65536   /* N6/8 */
#define NP4  16384   /* M5/8 */

#define BM 128
#define BN 64
#define BK 32

// ---------------- inverse-index construction ----------------
__global__ __launch_bounds__(256) void k_init_inv(int* inv5, int* inv4) {
    int t = blockIdx.x * 256 + threadIdx.x;
    if (t < M5) inv5[t] = -1;
    if (t < M4) inv4[t] = -1;
}

__global__ __launch_bounds__(256) void k_scatter_inv(const int* __restrict__ idx5,
                                                     const int* __restrict__ idx4,
                                                     int* inv5, int* inv4) {
    int t = blockIdx.x * 256 + threadIdx.x;
    if (t < NP5) inv5[idx5[t]] = t;
    if (t < NP4) inv4[idx4[t]] = t;
}

__global__ __launch_bounds__(256) void k_zero_stats(float* p) {
    p[blockIdx.x * 256 + threadIdx.x] = 0.0f;
}

// ---------------- fused double max-pool with octree padding ----------------
// Each 64-lane group handles one depth-4 row; float4 per lane, non-temporal
// streaming reads of `data` (read-once, 512MB > L2 -> NT keeps L2 for W/out4/H).
__global__ __launch_bounds__(256) void k_fused_pool(const float* __restrict__ data,
                                                    const int* __restrict__ inv5,
                                                    const int* __restrict__ inv4,
                                                    float* __restrict__ out4) {
    const int grp = threadIdx.x >> 6;                 // 4 rows per block
    const int r   = blockIdx.x * 4 + grp;
    const int c   = (threadIdx.x & 63) * 4;
    const int j   = inv4[r];
    f4v v = (f4v)0.0f;
    if (j >= 0) {
        v = (f4v)(-__builtin_inff());
        #pragma unroll
        for (int k = 0; k < 8; ++k) {
            const int i5 = inv5[j * 8 + k];
            if (i5 < 0) {
                v[0] = fmaxf(v[0], 0.0f); v[1] = fmaxf(v[1], 0.0f);
                v[2] = fmaxf(v[2], 0.0f); v[3] = fmaxf(v[3], 0.0f);
                continue;
            }
            const f4v* p = (const f4v*)(data + (size_t)i5 * 8 * CIN + c);
            #pragma unroll
            for (int m = 0; m < 8; ++m) {
                const f4v t = __builtin_nontemporal_load(p + m * (CIN / 4));
                v[0] = fmaxf(v[0], t[0]); v[1] = fmaxf(v[1], t[1]);
                v[2] = fmaxf(v[2], t[2]); v[3] = fmaxf(v[3], t[3]);
            }
        }
    }
    *(f4v*)(out4 + (size_t)r * CIN + c) = v;
}

// ---------------- WMMA f32 GEMM (H = A * W^T) + BN statistics ----------------
// Async global->LDS staging (double buffered, ASYNCcnt), K-pair-interleaved LDS
// layout so each WMMA fragment is one aligned ds_load_b64.
__global__ __launch_bounds__(256) void k_gemm_bnstats(const float* __restrict__ A,
                                                      const float* __restrict__ W,
                                                      float* __restrict__ H,
                                                      float* __restrict__ colsum,
                                                      float* __restrict__ colssq) {
    // layout: dword idx = ((k>>1)*BM + m)*2 + (k&1)  (A), same with BN for B
    __shared__ float ldsA[2][BK * BM];
    __shared__ float ldsB[2][BK * BN];
    __shared__ float bsum[BN];
    __shared__ float bssq[BN];

    const int tid   = threadIdx.x;
    const int mBase = blockIdx.x * BM;
    const int nBase = blockIdx.y * BN;

    if (tid < BN) { bsum[tid] = 0.0f; bssq[tid] = 0.0f; }

    const int lane = tid & 31;
    const int wv   = tid >> 5;        // 8 waves; wave wv owns M rows [wv*16, wv*16+16)
    const int half = lane >> 4;
    const int r    = lane & 15;
    const int mRow = wv * 16;

    const unsigned ldsA0 = (unsigned)(uintptr_t)&ldsA[0][0];
    const unsigned ldsA1 = (unsigned)(uintptr_t)&ldsA[1][0];
    const unsigned ldsB0 = (unsigned)(uintptr_t)&ldsB[0][0];
    const unsigned ldsB1 = (unsigned)(uintptr_t)&ldsB[1][0];

    v8f acc0 = {}, acc1 = {}, acc2 = {}, acc3 = {};

    // issue one K-tile of async global->LDS b64 copies (transposing, pair-interleaved)
    auto issue_tile = [&](int kk, int buf) {
        const unsigned la = buf ? ldsA1 : ldsA0;
        const unsigned lb = buf ? ldsB1 : ldsB0;
        #pragma unroll
        for (int q = 0; q < 8; ++q) {                 // A: 128 m x 16 k-pairs
            const int e = tid + q * 256;
            const int m = e >> 4;
            const int p = e & 15;
            const uint64_t ga = (uint64_t)A +
                (((uint64_t)(mBase + m) * CIN + (unsigned)(kk + 2 * p)) << 2);
            const unsigned da = la + (unsigned)((p * BM + m) * 8);
            asm volatile("global_load_async_to_lds_b64 %0, %1, off"
                         :: "v"(da), "v"(ga) : "memory");
        }
        #pragma unroll
        for (int q = 0; q < 4; ++q) {                 // B: 64 n x 16 k-pairs
            const int e = tid + q * 256;
            const int n = e >> 4;
            const int p = e & 15;
            const uint64_t ga = (uint64_t)W +
                (((uint64_t)(nBase + n) * CIN + (unsigned)(kk + 2 * p)) << 2);
            const unsigned db = lb + (unsigned)((p * BN + n) * 8);
            asm volatile("global_load_async_to_lds_b64 %0, %1, off"
                         :: "v"(db), "v"(ga) : "memory");
        }
    };

    issue_tile(0, 0);
    asm volatile("s_wait_asynccnt 0" ::: "memory");
    __syncthreads();

    for (int step = 0; step < CIN / BK; ++step) {
        const int cur = step & 1;
        if (step < CIN / BK - 1) issue_tile((step + 1) * BK, cur ^ 1);

        const float* As = &ldsA[cur][0];
        const float* Bs = &ldsB[cur][0];
        #pragma unroll
        for (int k0 = 0; k0 < BK; k0 += 4) {
            const int p0 = (k0 >> 1) + half;          // K-pair index: K = 2*p0 + vgpr
            const v2f a  = *(const v2f*)(As + (p0 * BM + mRow + r) * 2);
            const v2f b0 = *(const v2f*)(Bs + (p0 * BN +  0 + r) * 2);
            const v2f b1 = *(const v2f*)(Bs + (p0 * BN + 16 + r) * 2);
            const v2f b2 = *(const v2f*)(Bs + (p0 * BN + 32 + r) * 2);
            const v2f b3 = *(const v2f*)(Bs + (p0 * BN + 48 + r) * 2);
            acc0 = __builtin_amdgcn_wmma_f32_16x16x4_f32(false, a, false, b0, (short)0, acc0, false, false);
            acc1 = __builtin_amdgcn_wmma_f32_16x16x4_f32(false, a, false, b1, (short)0, acc1, false, false);
            acc2 = __builtin_amdgcn_wmma_f32_16x16x4_f32(false, a, false, b2, (short)0, acc2, false, false);
            acc3 = __builtin_amdgcn_wmma_f32_16x16x4_f32(false, a, false, b3, (short)0, acc3, false, false);
        }
        asm volatile("s_wait_asynccnt 0" ::: "memory");
        __syncthreads();
    }

    // epilogue: store H, reduce per-column sum / sum-of-squares
    v8f accs[4] = {acc0, acc1, acc2, acc3};
    #pragma unroll
    for (int nt = 0; nt < 4; ++nt) {
        float s = 0.0f, ss = 0.0f;
        const int col = nBase + nt * 16 + r;
        #pragma unroll
        for (int e = 0; e < 8; ++e) {
            const float val = accs[nt][e];             // D: M = 8*half + e, N = r
            H[(size_t)(mBase + mRow + half * 8 + e) * COUT + col] = val;
            s  += val;
            ss += val * val;
        }
        atomicAdd(&bsum[nt * 16 + r], s);
        atomicAdd(&bssq[nt * 16 + r], ss);
    }
    __syncthreads();
    if (tid < BN) {
        atomicAdd(&colsum[nBase + tid], bsum[tid]);
        atomicAdd(&colssq[nBase + tid], bssq[tid]);
    }
}

// ---------------- BN parameter fold ----------------
__global__ __launch_bounds__(512) void k_bn_params(const float* __restrict__ colsum,
                                                   const float* __restrict__ colssq,
                                                   const float* __restrict__ gamma,
                                                   const float* __restrict__ beta,
                                                   float* __restrict__ scale,
                                                   float* __restrict__ shift) {
    const int c = threadIdx.x;
    const float inv_n = 1.0f / (float)M4;
    const float mean = colsum[c] * inv_n;
    const float var  = colssq[c] * inv_n - mean * mean;
    const float sc   = gamma[c] * rsqrtf(var + 1e-5f);
    scale[c] = sc;
    shift[c] = beta[c] - mean * sc;
}

// ---------------- BN apply + ReLU (float4, NT output store) ----------------
__global__ __launch_bounds__(256) void k_bn_relu(const float* __restrict__ H,
                                                 const float* __restrict__ scale,
                                                 const float* __restrict__ shift,
                                                 float* __restrict__ out) {
    const size_t i = ((size_t)blockIdx.x * 256 + threadIdx.x) * 4;
    const int c = (int)(i & (COUT - 1));
    const f4v h4 = *(const f4v*)(H + i);
    const f4v sc = *(const f4v*)(scale + c);
    const f4v sh = *(const f4v*)(shift + c);
    f4v v;
    v[0] = fmaxf(fmaf(h4[0], sc[0], sh[0]), 0.0f);
    v[1] = fmaxf(fmaf(h4[1], sc[1], sh[1]), 0.0f);
    v[2] = fmaxf(fmaf(h4[2], sc[2], sh[2]), 0.0f);
    v[3] = fmaxf(fmaf(h4[3], sc[3], sh[3]), 0.0f);
    __builtin_nontemporal_store(v, (f4v*)(out + i));
}

extern "C" void kernel_launch(void* const* d_in, const int* in_sizes, int n_in,
                              void* d_out, int out_size, void* d_ws, size_t ws_size,
                              hipStream_t stream) {
    const float* data   = (const float*)d_in[0];
    const float* weight = (const float*)d_in[1];
    const float* gamma  = (const float*)d_in[2];
    const float* beta   = (const float*)d_in[3];
    const int*   idx5   = (const int*)d_in[4];
    const int*   idx4   = (const int*)d_in[5];

    char* ws = (char*)d_ws;
    int*   inv5   = (int*)(ws);                                   // 512 KB
    int*   inv4   = (int*)(ws + 524288);                          // 128 KB
    float* out4   = (float*)(ws + (1u << 20));                    // 32 MB
    float* H      = (float*)(ws + (1u << 20) + 33554432u);        // 64 MB
    float* stats  = (float*)(ws + (1u << 20) + 33554432u + 67108864u);
    float* colsum = stats;
    float* colssq = stats + 512;
    float* scale  = stats + 1024;
    float* shift  = stats + 1536;
    float* out    = (float*)d_out;

    k_init_inv   <<<512, 256, 0, stream>>>(inv5, inv4);
    k_scatter_inv<<<256, 256, 0, stream>>>(idx5, idx4, inv5, inv4);
    k_zero_stats <<<4,   256, 0, stream>>>(stats);
    k_fused_pool <<<M4 / 4, 256, 0, stream>>>(data, inv5, inv4, out4);

    dim3 g(M4 / BM, COUT / BN);
    k_gemm_bnstats<<<g, 256, 0, stream>>>(out4, weight, H, colsum, colssq);
    k_bn_params  <<<1, 512, 0, stream>>>(colsum, colssq, gamma, beta, scale, shift);
    k_bn_relu    <<<(M4 * COUT) / 1024, 256, 0, stream>>>(H, scale, shift, out);
}